// LinearLayer_34995393527975
// MI455X (gfx1250) — compile-verified
//
#include <hip/hip_runtime.h>
#include <hip/hip_bf16.h>

// hidden = tanh(H @ h + C @ c); comm = hidden.
// h,c: [64, D] f32 (D = 524288 fixed), H,C: [64,64] f32.
// d_out: [2, 64, D] f32 (hidden then comm).
//
// Memory-bound: ~537 MB total traffic @ 23.3 TB/s => ~23us floor.
// Each wave owns one 16-column tile and computes all 64 output rows
// (4 row-tiles) with V_WMMA_F32_16X16X4_F32, so streamed B fragments are
// reused 4x from registers and h/c are read from HBM exactly once.
// H/C staged once per block in bank-conflict-padded LDS.

typedef __attribute__((ext_vector_type(2))) float v2f;
typedef __attribute__((ext_vector_type(8))) float v8f;

#define NAGENTS 64
constexpr long FEAT_DIM = 524288;  // compile-time: folds strides into IOFFSET/SGPR bases
#define LDS_STRIDE 68              // 64 + 4 pad: b64 A-frag reads hit distinct bank pairs

__device__ __forceinline__ float fast_tanhf(float x) {
  // tanh(x) = (e^2x - 1) / (e^2x + 1); v_exp_f32 + v_rcp_f32 path.
  float e = __expf(2.0f * x);
  return __fdividef(e - 1.0f, e + 1.0f);
}

__global__ __launch_bounds__(256) void comm_gemm_tanh_kernel(
    const float* __restrict__ h,   // [64, D]
    const float* __restrict__ c,   // [64, D]
    const float* __restrict__ Hm,  // [64, 64]
    const float* __restrict__ Cm,  // [64, 64]
    float* __restrict__ out) {     // [2, 64, D]
  constexpr long D = FEAT_DIM;
  __shared__ float sH[NAGENTS * LDS_STRIDE];
  __shared__ float sC[NAGENTS * LDS_STRIDE];

  const int tid = threadIdx.x;

  // Cooperative load of H and C into padded LDS (4096 elems each, 256 thr).
  #pragma unroll
  for (int i = 0; i < (NAGENTS * NAGENTS) / 256; ++i) {
    int idx = tid + i * 256;
    int row = idx >> 6;
    int k   = idx & 63;
    sH[row * LDS_STRIDE + k] = Hm[idx];
    sC[row * LDS_STRIDE + k] = Cm[idx];
  }
  __syncthreads();

  const int wave = tid >> 5;         // 8 waves per block
  const int lane = tid & 31;
  const int n    = lane & 15;        // output column within tile / A-row M%16
  const int kh   = (lane >> 4) << 1; // K sub-offset: 0 (lanes 0-15), 2 (lanes 16-31)

  const long col0 = ((long)blockIdx.x * 8 + wave) * 16;  // 16-col tile base

  // Per-lane streaming pointers at row kh, column col0+n; advance by 4 rows/step.
  const float* __restrict__ hp = h + (long)kh * D + col0 + n;
  const float* __restrict__ cp = c + (long)kh * D + col0 + n;

  v8f acc0 = {}, acc1 = {}, acc2 = {}, acc3 = {};

  #pragma unroll
  for (int kb = 0; kb < NAGENTS; kb += 4) {
    const int k0 = kb + kh;
    // B fragments (4x16 f32): lane holds B[k0][n] and B[k0+1][n].
    // Row stride D*4B = 2 MiB fits the signed 24-bit instruction offset.
    v2f bh, bc;
    bh.x = hp[0];
    bh.y = hp[D];
    bc.x = cp[0];
    bc.y = cp[D];

    if (kb + 4 < NAGENTS) {
      __builtin_prefetch(hp + 4 * D, 0, 0);  // global_prefetch_b8
      __builtin_prefetch(cp + 4 * D, 0, 0);
    }
    hp += 4 * D;
    cp += 4 * D;

    // A fragments (16x4 f32) from LDS: lane holds A[n + 16*rt][k0..k0+1].
    {
      v2f aH = *(const v2f*)&sH[(n + 0) * LDS_STRIDE + k0];
      v2f aC = *(const v2f*)&sC[(n + 0) * LDS_STRIDE + k0];
      acc0 = __builtin_amdgcn_wmma_f32_16x16x4_f32(false, aH, false, bh,
                                                   (short)0, acc0, false, false);
      acc0 = __builtin_amdgcn_wmma_f32_16x16x4_f32(false, aC, false, bc,
                                                   (short)0, acc0, false, false);
    }
    {
      v2f aH = *(const v2f*)&sH[(n + 16) * LDS_STRIDE + k0];
      v2f aC = *(const v2f*)&sC[(n + 16) * LDS_STRIDE + k0];
      acc1 = __builtin_amdgcn_wmma_f32_16x16x4_f32(false, aH, false, bh,
                                                   (short)0, acc1, false, false);
      acc1 = __builtin_amdgcn_wmma_f32_16x16x4_f32(false, aC, false, bc,
                                                   (short)0, acc1, false, false);
    }
    {
      v2f aH = *(const v2f*)&sH[(n + 32) * LDS_STRIDE + k0];
      v2f aC = *(const v2f*)&sC[(n + 32) * LDS_STRIDE + k0];
      acc2 = __builtin_amdgcn_wmma_f32_16x16x4_f32(false, aH, false, bh,
                                                   (short)0, acc2, false, false);
      acc2 = __builtin_amdgcn_wmma_f32_16x16x4_f32(false, aC, false, bc,
                                                   (short)0, acc2, false, false);
    }
    {
      v2f aH = *(const v2f*)&sH[(n + 48) * LDS_STRIDE + k0];
      v2f aC = *(const v2f*)&sC[(n + 48) * LDS_STRIDE + k0];
      acc3 = __builtin_amdgcn_wmma_f32_16x16x4_f32(false, aH, false, bh,
                                                   (short)0, acc3, false, false);
      acc3 = __builtin_amdgcn_wmma_f32_16x16x4_f32(false, aC, false, bc,
                                                   (short)0, acc3, false, false);
    }
  }

  // Epilogue: fast tanh + store to hidden and comm (identical halves).
  const int mrow = (lane >> 4) << 3;  // 0 or 8: C/D row base within tile
  v8f accs[4] = {acc0, acc1, acc2, acc3};

  // Row-walking store pointers: one pointer bump (+D) per row, compile-time
  // strides let the compiler keep bases in SGPRs (GVS addressing).
  float* __restrict__ oH = out + (long)mrow * D + col0 + n;
  float* __restrict__ oC = oH + (long)NAGENTS * D;

  #pragma unroll
  for (int rt = 0; rt < 4; ++rt) {
    #pragma unroll
    for (int v = 0; v < 8; ++v) {
      const float t = fast_tanhf(accs[rt][v]);
      const long off = ((long)rt * 16 + v) * D;
      oH[off] = t;
      oC[off] = t;
    }
  }
}

extern "C" void kernel_launch(void* const* d_in, const int* in_sizes, int n_in,
                              void* d_out, int out_size, void* d_ws, size_t ws_size,
                              hipStream_t stream) {
  const float* h  = (const float*)d_in[0];
  const float* c  = (const float*)d_in[1];
  const float* Hm = (const float*)d_in[2];
  const float* Cm = (const float*)d_in[3];
  float* out = (float*)d_out;

  const int blocks = (int)(FEAT_DIM / (16 * 8));  // one 16-col tile per wave, 8 waves/block

  comm_gemm_tanh_kernel<<<blocks, 256, 0, stream>>>(h, c, Hm, Cm, out);
}